// STCell_76587856822321
// MI455X (gfx1250) — compile-verified
//
#include <hip/hip_runtime.h>
#include <hip/hip_bf16.h>

#define NN      8192          // 2*N nodes
#define SEGOFF  4096          // init_index*N
#define ROWS_X  64            // B*C_FEAT
#define KT      64            // K stage width (double-buffered)
#define LSTR    68            // padded LDS row stride (floats): mult of 4, 68%64=4

typedef __attribute__((ext_vector_type(2))) float v2f;
typedef __attribute__((ext_vector_type(8))) float v8f;
typedef __attribute__((ext_vector_type(4))) int   v4i;

// Async global->LDS path (CDNA5 GLOBAL_LOAD_ASYNC_TO_LDS_B128, ASYNCcnt).
#if defined(__AMDGCN__) && __has_builtin(__builtin_amdgcn_global_load_async_to_lds_b128) && __has_builtin(__builtin_amdgcn_s_wait_asynccnt)
#define HOP_ASYNC 1
#else
#define HOP_ASYNC 0
#endif

#if HOP_ASYNC
typedef __attribute__((address_space(1))) v4i gv4i;   // global b128 payload
typedef __attribute__((address_space(3))) v4i lv4i;   // LDS b128 payload
__device__ __forceinline__ void async_ld_b128(const float* g, float* l) {
    __builtin_amdgcn_global_load_async_to_lds_b128((gv4i*)g, (lv4i*)l, 0, 0);
}
#endif

// ---------------------------------------------------------------------------
// Kernel 1: adj_raw[v,w] = s1a*af0 + s1b*af1 + s2a*as0 + s2b*as1 + mask,
// plus inv_rowsum[v].  One block per row (fully coalesced float4 streams).
// Row-normalization is NOT applied to adj: it commutes with the matmul and
// is applied as a per-row output scale in the hop kernel.
// ---------------------------------------------------------------------------
__global__ void __launch_bounds__(256)
build_adj(const float* __restrict__ af, const float* __restrict__ as,
          const float* __restrict__ am, const float* __restrict__ a1p,
          const float* __restrict__ a2p, float* __restrict__ adj,
          float* __restrict__ inv_rs) {
    const int v   = blockIdx.x;
    const int tid = threadIdx.x;
    // softmax of the two 2-element alpha vectors (cheap, L2-cached)
    float a0 = a1p[0], a1 = a1p[1];
    float m  = fmaxf(a0, a1);
    float e0 = __expf(a0 - m), e1 = __expf(a1 - m);
    const float s10 = e0 / (e0 + e1), s11 = e1 / (e0 + e1);
    a0 = a2p[0]; a1 = a2p[1];
    m  = fmaxf(a0, a1);
    e0 = __expf(a0 - m); e1 = __expf(a1 - m);
    const float s20 = e0 / (e0 + e1), s21 = e1 / (e0 + e1);

    const size_t row   = (size_t)v * NN;
    const size_t plane = (size_t)NN * NN;
    const float4* f0 = (const float4*)(af + row);
    const float4* f1 = (const float4*)(af + plane + row);
    const float4* g0 = (const float4*)(as + row);
    const float4* g1 = (const float4*)(as + plane + row);
    const float4* mm = (const float4*)(am + row);
    float4* outp     = (float4*)(adj + row);

    float acc = 0.f;
    #pragma unroll
    for (int j = 0; j < 8; ++j) {              // 8192/4 float4s / 256 thr = 8
        const int i = tid + j * 256;
        float4 x0 = f0[i], x1 = f1[i], y0 = g0[i], y1 = g1[i], z = mm[i];
        float4 r;
        r.x = s10*x0.x + s11*x1.x + s20*y0.x + s21*y1.x + z.x;
        r.y = s10*x0.y + s11*x1.y + s20*y0.y + s21*y1.y + z.y;
        r.z = s10*x0.z + s11*x1.z + s20*y0.z + s21*y1.z + z.z;
        r.w = s10*x0.w + s11*x1.w + s20*y0.w + s21*y1.w + z.w;
        outp[i] = r;
        acc += (r.x + r.y) + (r.z + r.w);
    }
    __shared__ float red[256];
    red[tid] = acc;
    __syncthreads();
    for (int s = 128; s > 0; s >>= 1) {
        if (tid < s) red[tid] += red[tid + s];
        __syncthreads();
    }
    if (tid == 0) inv_rs[v] = 1.0f / red[0];
}

// ---------------------------------------------------------------------------
// Kernel 2: one hop, Y[bc, v] = inv_rs[v] * sum_w adj[v,w] * X[bc, w]
// via V_WMMA_F32_16X16X4_F32.  Block = 16-row M-tile, 4 waves = 4 N-tiles
// over the 64 bc columns; K streamed in 64-wide LDS stages, double-buffered
// with async global->LDS copies (ASYNCcnt pipelining) when available.
// fp32 WMMA fragment mapping (ISA 7.12.2):
//   A (16x4): lane<16 -> {K=0,K=1} of row M=lane; lane>=16 -> {K=2,K=3}
//   B (4x16): VGPR0 = row K=(lane>=16?2:0) striped over N=lane&15, VGPR1 = K+1
//   D (16x16): VGPR i -> M = i + 8*(lane>=16), N = lane&15
// ---------------------------------------------------------------------------
__global__ void __launch_bounds__(128)
hop_wmma(const float* __restrict__ adj, const float* __restrict__ inv_rs,
         const float* __restrict__ X, int xstride, float* __restrict__ Y) {
    __shared__ float sA[2][16][LSTR];
    __shared__ float sX[2][ROWS_X][LSTR];

    const int tid  = threadIdx.x;
    const int wave = tid >> 5;
    const int lane = tid & 31;
    const int v0   = blockIdx.x * 16;
    const int n0   = wave * 16;

    const int rA = lane & 15;
    const int kh = (lane >> 4) << 1;     // 0 or 2
    const int rB = n0 + (lane & 15);

    v8f c = {};

#if HOP_ASYNC
    // Issue one K-stage: adj tile 16x64 (256 b128) + X tile 64x64 (1024 b128)
    // = 10 async instructions per thread -> ASYNCcnt += 10 per wave per stage.
    auto stage = [&](int buf, int k0) {
        #pragma unroll
        for (int j = 0; j < 2; ++j) {
            const int idx = tid + j * 128;
            const int r = idx >> 4, c4 = idx & 15;
            async_ld_b128(adj + (size_t)(v0 + r) * NN + k0 + c4 * 4,
                          &sA[buf][r][c4 * 4]);
        }
        #pragma unroll
        for (int j = 0; j < 8; ++j) {
            const int idx = tid + j * 128;
            const int r = idx >> 4, c4 = idx & 15;
            async_ld_b128(X + (size_t)r * xstride + k0 + c4 * 4,
                          &sX[buf][r][c4 * 4]);
        }
    };

    stage(0, 0);
    int buf = 0;
    for (int k0 = 0; k0 < NN; k0 += KT) {
        if (k0 + KT < NN) {
            stage(buf ^ 1, k0 + KT);                 // prefetch next stage
            __builtin_amdgcn_s_wait_asynccnt(10);    // drain current stage only
        } else {
            __builtin_amdgcn_s_wait_asynccnt(0);
        }
        __syncthreads();                             // all waves' tiles landed
        #pragma unroll
        for (int kk = 0; kk < KT; kk += 4) {
            v2f a = *(const v2f*)&sA[buf][rA][kk + kh];
            v2f b = *(const v2f*)&sX[buf][rB][kk + kh];
            c = __builtin_amdgcn_wmma_f32_16x16x4_f32(
                    false, a, false, b, (short)0, c, false, false);
        }
        __syncthreads();                             // readers done before reuse
        buf ^= 1;
    }
#else
    // Synchronous staging fallback (global->VGPR->LDS), single buffer.
    for (int k0 = 0; k0 < NN; k0 += KT) {
        #pragma unroll
        for (int j = 0; j < 2; ++j) {
            const int idx = tid + j * 128;
            const int r = idx >> 4, c4 = idx & 15;
            float4 t = *(const float4*)(adj + (size_t)(v0 + r) * NN + k0 + c4 * 4);
            *(float4*)&sA[0][r][c4 * 4] = t;
        }
        #pragma unroll
        for (int j = 0; j < 8; ++j) {
            const int idx = tid + j * 128;
            const int r = idx >> 4, c4 = idx & 15;
            float4 t = *(const float4*)(X + (size_t)r * xstride + k0 + c4 * 4);
            *(float4*)&sX[0][r][c4 * 4] = t;
        }
        __syncthreads();
        if (k0 + KT < NN)
            __builtin_prefetch(adj + (size_t)(v0 + (tid >> 3)) * NN + k0 + KT + (tid & 7) * 8, 0, 1);
        #pragma unroll
        for (int kk = 0; kk < KT; kk += 4) {
            v2f a = *(const v2f*)&sA[0][rA][kk + kh];
            v2f b = *(const v2f*)&sX[0][rB][kk + kh];
            c = __builtin_amdgcn_wmma_f32_16x16x4_f32(
                    false, a, false, b, (short)0, c, false, false);
        }
        __syncthreads();
    }
#endif

    // scale by 1/rowsum on the way out (normalization commutes with matmul)
    const int nl = lane & 15;
    const int mh = (lane >> 4) << 3;
    #pragma unroll
    for (int i = 0; i < 8; ++i) {
        const int v = v0 + mh + i;
        Y[(size_t)(n0 + nl) * NN + v] = c[i] * inv_rs[v];
    }
}

// ---------------------------------------------------------------------------
// Kernel 3: y = W(128x16) @ seg + b, store y, accumulate per-channel sum/sumsq
// ---------------------------------------------------------------------------
__global__ void __launch_bounds__(128)
glu_pass1(const float* __restrict__ Xhop, const float* __restrict__ W,
          const float* __restrict__ bias, float* __restrict__ ybuf,
          float* __restrict__ stats) {
    __shared__ float sSeg[16][129];
    const int o  = threadIdx.x;          // output channel 0..127
    const int b  = blockIdx.y;           // batch 0..3
    const int n0 = blockIdx.x * 128;     // 32 chunks cover 4096

    float w[16];
    #pragma unroll
    for (int cc = 0; cc < 16; ++cc) w[cc] = W[o * 16 + cc];
    const float bb = bias[o];

    #pragma unroll
    for (int j = 0; j < 16; ++j) {
        const int idx = threadIdx.x + j * 128;
        const int cch = idx >> 7;
        const int nn  = idx & 127;
        sSeg[cch][nn] = Xhop[(size_t)(b * 16 + cch) * NN + SEGOFF + n0 + nn];
    }
    __syncthreads();

    float s = 0.f, sq = 0.f;
    for (int nn = 0; nn < 128; ++nn) {
        float y = bb;
        #pragma unroll
        for (int cc = 0; cc < 16; ++cc) y = fmaf(w[cc], sSeg[cc][nn], y);
        ybuf[(size_t)(b * 128 + o) * SEGOFF + n0 + nn] = y;
        s += y; sq += y * y;
    }
    atomicAdd(&stats[o],       s);
    atomicAdd(&stats[128 + o], sq);
}

// Kernel 4: finalize BN scale/shift per channel (population variance)
__global__ void glu_stats(const float* __restrict__ stats,
                          const float* __restrict__ gamma,
                          const float* __restrict__ beta,
                          float* __restrict__ ss) {
    const int o = threadIdx.x;           // 128 threads
    const float inv_n = 1.0f / 16384.0f; // B * N samples per channel
    const float mean = stats[o] * inv_n;
    const float var  = stats[128 + o] * inv_n - mean * mean;
    const float sc   = gamma[o] * rsqrtf(var + 1e-5f);
    ss[o]       = sc;
    ss[128 + o] = beta[o] - mean * sc;
}

// Kernel 5: BN + GLU + running max into d_out
__global__ void __launch_bounds__(256)
glu_apply(const float* __restrict__ ybuf, const float* __restrict__ ss,
          float* __restrict__ out, int first) {
    const int idx = blockIdx.x * 256 + threadIdx.x;   // 4*64*4096 total
    const int n = idx & 4095;
    const int o = (idx >> 12) & 63;
    const int b = idx >> 18;
    const float l = ybuf[(size_t)(b * 128 + o)      * SEGOFF + n] * ss[o]      + ss[128 + o];
    const float r = ybuf[(size_t)(b * 128 + o + 64) * SEGOFF + n] * ss[o + 64] + ss[192 + o];
    float val = l * (1.0f / (1.0f + __expf(-r)));
    if (!first) val = fmaxf(val, out[idx]);
    out[idx] = val;
}

// ---------------------------------------------------------------------------
extern "C" void kernel_launch(void* const* d_in, const int* in_sizes, int n_in,
                              void* d_out, int out_size, void* d_ws, size_t ws_size,
                              hipStream_t stream) {
    const float* adj_first  = (const float*)d_in[0];
    const float* adj_second = (const float*)d_in[1];
    const float* adj_mask   = (const float*)d_in[2];
    const float* data       = (const float*)d_in[3];
    const float* alpha1     = (const float*)d_in[4];
    const float* alpha2     = (const float*)d_in[5];
    const float* glu_w      = (const float*)d_in[6];
    const float* glu_b      = (const float*)d_in[7];
    const float* bn_gamma   = (const float*)d_in[8];
    const float* bn_beta    = (const float*)d_in[9];
    float* out = (float*)d_out;

    // workspace layout
    float* adj    = (float*)d_ws;                       // 8192*8192
    float* inv_rs = adj    + (size_t)NN * NN;           // 8192
    float* xbuf0  = inv_rs + NN;                        // 64*8192
    float* xbuf1  = xbuf0  + (size_t)ROWS_X * NN;       // 64*8192
    float* ybuf   = xbuf1  + (size_t)ROWS_X * NN;       // 4*128*4096
    float* stats  = ybuf   + (size_t)4 * 128 * SEGOFF;  // 256
    float* ss     = stats  + 256;                       // 256

    build_adj<<<NN, 256, 0, stream>>>(adj_first, adj_second, adj_mask,
                                      alpha1, alpha2, adj, inv_rs);

    const float* Xcur = data;        // data_long == data[:,:,0:8192] (DILATED=1)
    int xstride = 16384;
    float* xb[2] = { xbuf0, xbuf1 };

    for (int hop = 0; hop < 3; ++hop) {
        float* Ynext = xb[hop & 1];
        hop_wmma<<<NN / 16, 128, 0, stream>>>(adj, inv_rs, Xcur, xstride, Ynext);

        (void)hipMemsetAsync(stats, 0, 256 * sizeof(float), stream);
        glu_pass1<<<dim3(32, 4), 128, 0, stream>>>(Ynext,
                                                   glu_w + (size_t)hop * 128 * 16,
                                                   glu_b + (size_t)hop * 128,
                                                   ybuf, stats);
        glu_stats<<<1, 128, 0, stream>>>(stats,
                                         bn_gamma + (size_t)hop * 128,
                                         bn_beta  + (size_t)hop * 128, ss);
        glu_apply<<<(4 * 64 * SEGOFF) / 256, 256, 0, stream>>>(ybuf, ss, out, hop == 0);

        Xcur = Ynext;
        xstride = NN;
    }
}